// KSpaceMap_87900800680502
// MI455X (gfx1250) — compile-verified
//
#include <hip/hip_runtime.h>

typedef __attribute__((ext_vector_type(2))) float v2f;
typedef __attribute__((ext_vector_type(4))) float v4f;
typedef __attribute__((ext_vector_type(8))) float v8f;

#define TWO_PI_OVER_128 0.04908738521234052f  // 2*pi/128

// ---------------------------------------------------------------------------
// Phase 1: per-batch fp32 DFT via WMMA.
//   Ar[b,h,i] = sum_w x[b,h,w] * cos(2*pi*w*i/128)
//   Ai[b,h,i] = sum_w x[b,h,w] * (-sin(2*pi*w*i/128))   (== Im of fft)
// Stored transposed: art[b][i][h], ait[b][i][h]  (so phase 2 reads contiguous)
// One block per batch, 8 waves; wave ti owns output tile-row ti (16 h values).
// ---------------------------------------------------------------------------
__global__ __launch_bounds__(256) void dft_wmma_kernel(
    const float* __restrict__ x, float* __restrict__ art, float* __restrict__ ait) {
  __shared__ float ct[128];   // cos(2*pi*t/128)
  __shared__ float stn[128];  // -sin(2*pi*t/128)
  const int tid = threadIdx.x;
  if (tid < 128) {
    float ang = TWO_PI_OVER_128 * (float)tid;
    ct[tid]  =  __cosf(ang);
    stn[tid] = -__sinf(ang);
  }
  __syncthreads();

  const int b    = blockIdx.x;
  const int lane = tid & 31;
  const int ti   = tid >> 5;      // wave id = output tile row (0..7)
  const int lhi  = lane >> 4;     // 0: K={0,1}, 1: K={2,3} within a WMMA step
  const int llo  = lane & 15;

  const float* xb = x + (size_t)b * (128 * 128);

  // Preload all 32 A-fragments for this tile row (A = 16x4 fp32 per step).
  // Lane layout: afrag.x = X[h][kbase], afrag.y = X[h][kbase+1],
  //   h = ti*16 + llo, kbase = 4k + 2*lhi.
  v2f afrag[32];
  {
    const float* base = xb + (size_t)(ti * 16 + llo) * 128 + lhi * 2;
#pragma unroll
    for (int k = 0; k < 32; ++k) afrag[k] = *(const v2f*)(base + k * 4);
  }

  for (int tj = 0; tj < 8; ++tj) {
    const int n = tj * 16 + llo;  // output column (frequency i)
    v8f accr = {0.f, 0.f, 0.f, 0.f, 0.f, 0.f, 0.f, 0.f};
    v8f acci = {0.f, 0.f, 0.f, 0.f, 0.f, 0.f, 0.f, 0.f};
#pragma unroll
    for (int k = 0; k < 32; ++k) {
      // B (4x16) lane layout: b.x = M[kbase][n], b.y = M[kbase+1][n]
      const int kbase = k * 4 + lhi * 2;
      const int t0 = (kbase * n) & 127;
      const int t1 = (t0 + n) & 127;
      v2f bc, bs;
      bc.x = ct[t0];  bc.y = ct[t1];
      bs.x = stn[t0]; bs.y = stn[t1];
      accr = __builtin_amdgcn_wmma_f32_16x16x4_f32(
          false, afrag[k], false, bc, (short)0, accr, false, false);
      acci = __builtin_amdgcn_wmma_f32_16x16x4_f32(
          false, afrag[k], false, bs, (short)0, acci, false, false);
    }
    // D layout: vgpr r, lanes 0-15 -> M=r; lanes 16-31 -> M=8+r.
    // h = ti*16 + lhi*8 + r  (r=0..7 contiguous) -> two b128 stores, transposed.
    const size_t base = (size_t)b * 16384 + (size_t)n * 128 + ti * 16 + lhi * 8;
    v4f r0 = {accr[0], accr[1], accr[2], accr[3]};
    v4f r1 = {accr[4], accr[5], accr[6], accr[7]};
    v4f i0 = {acci[0], acci[1], acci[2], acci[3]};
    v4f i1 = {acci[4], acci[5], acci[6], acci[7]};
    *(v4f*)(art + base)     = r0;
    *(v4f*)(art + base + 4) = r1;
    *(v4f*)(ait + base)     = i0;
    *(v4f*)(ait + base + 4) = i1;
  }
}

// ---------------------------------------------------------------------------
// Phase 2: streaming expansion (store-bandwidth bound, 256 MB).
// One block per (b,i) plane. out[b,i,h,w] = (Ar*cos(2pi*i*w/128) - Ai*sin)/128.
// Thread owns fixed w-quad -> trig factors are loop-invariant registers.
// Per wave-instruction: 32 lanes store one contiguous 512B row, nontemporal.
// ---------------------------------------------------------------------------
__global__ __launch_bounds__(256) void expand_kernel(
    const float* __restrict__ art, const float* __restrict__ ait,
    float* __restrict__ out) {
  __shared__ float sAr[128], sAi[128], ctb[128], stb[128];
  const int i = blockIdx.x;
  const int b = blockIdx.y;
  const int tid = threadIdx.x;

  const size_t abase = (size_t)b * 16384 + (size_t)i * 128;
  if (tid < 128) {
    sAr[tid] = art[abase + tid];
    sAi[tid] = ait[abase + tid];
  } else {
    const int t = tid - 128;
    float ang = TWO_PI_OVER_128 * (float)t;
    ctb[t] = __cosf(ang) * (1.0f / 128.0f);
    stb[t] = __sinf(ang) * (1.0f / 128.0f);
  }
  __syncthreads();

  const int w0 = (tid & 31) * 4;   // lane -> contiguous w quad
  const int h0 = tid >> 5;         // wave -> h phase (0..7)
  const float c0 = ctb[(i * w0) & 127];
  const float c1 = ctb[(i * (w0 + 1)) & 127];
  const float c2 = ctb[(i * (w0 + 2)) & 127];
  const float c3 = ctb[(i * (w0 + 3)) & 127];
  const float s0 = stb[(i * w0) & 127];
  const float s1 = stb[(i * (w0 + 1)) & 127];
  const float s2 = stb[(i * (w0 + 2)) & 127];
  const float s3 = stb[(i * (w0 + 3)) & 127];

  float* ob = out + ((size_t)(b * 128 + i) * 16384);
#pragma unroll
  for (int h = h0; h < 128; h += 8) {
    const float ar = sAr[h];  // LDS broadcast (all lanes same h)
    const float ai = sAi[h];
    v4f v;
    v.x = ar * c0 - ai * s0;
    v.y = ar * c1 - ai * s1;
    v.z = ar * c2 - ai * s2;
    v.w = ar * c3 - ai * s3;
    __builtin_nontemporal_store(v, (v4f*)(ob + h * 128 + w0));
  }
}

extern "C" void kernel_launch(void* const* d_in, const int* in_sizes, int n_in,
                              void* d_out, int out_size, void* d_ws, size_t ws_size,
                              hipStream_t stream) {
  const float* x = (const float*)d_in[0];
  // d_in[1] (mask) is not used by the reference output.
  float* art = (float*)d_ws;                 // 32*128*128 floats (2 MB)
  float* ait = art + 32 * 128 * 128;         // 2 MB more
  dft_wmma_kernel<<<dim3(32), dim3(256), 0, stream>>>(x, art, ait);
  expand_kernel<<<dim3(128, 32), dim3(256), 0, stream>>>(art, ait, (float*)d_out);
}